// SupConLoss_40226663694799
// MI455X (gfx1250) — compile-verified
//
#include <hip/hip_runtime.h>
#include <hip/hip_bf16.h>
#include <math.h>

typedef __attribute__((ext_vector_type(2))) float v2f;
typedef __attribute__((ext_vector_type(8))) float v8f;

#define BATCH 32
#define SEQ   1024
#define DIM   768
#define LDA   770   // padded LDS row stride (770 % 64 == 2 -> conflict-free b64 reads)

__global__ void zero_out_kernel(float* out) { out[0] = 0.0f; }

// Per-batch: group_size[b,i] = #(j : label[b,j] == label[b,i])  (labels in [0,16))
__global__ __launch_bounds__(256) void label_stats_kernel(const int* __restrict__ labels,
                                                          int* __restrict__ gsize) {
    __shared__ int labs[SEQ];
    __shared__ int hist[16];
    const int b = blockIdx.x, tid = threadIdx.x;
    if (tid < 16) hist[tid] = 0;
    for (int i = tid; i < SEQ; i += 256) labs[i] = labels[b * SEQ + i];
    __syncthreads();
    for (int i = tid; i < SEQ; i += 256) atomicAdd(&hist[labs[i] & 15], 1);
    __syncthreads();
    for (int i = tid; i < SEQ; i += 256) gsize[b * SEQ + i] = hist[labs[i] & 15];
}

// One block per (batch, 16-row anchor tile). 4 waves, each wave owns a 16x16 WMMA tile
// and strides over j-tiles. A rows staged in LDS; B columns read from global (L2-resident).
__global__ __launch_bounds__(128) void supcon_main_kernel(const float* __restrict__ feat,
                                                          const int* __restrict__ labels,
                                                          const int* __restrict__ gsize,
                                                          float* __restrict__ out) {
    __shared__ float As[16 * LDA];
    __shared__ int   labsh[SEQ];
    __shared__ float rowInv[16];                 // 1 / max(S - |P|, 1) per anchor row
    __shared__ float negS[16], posS[16], nLat[16];

    const int b     = blockIdx.y;
    const int iBase = blockIdx.x * 16;
    const int tid   = threadIdx.x;
    const int wave  = tid >> 5;
    const int lane  = tid & 31;
    const int n     = lane & 15;   // column within tile (and A row m, same layout)
    const int hi    = lane >> 4;   // half-wave select: K offset +2, row offset +8 for C

    // Stage labels for the whole batch row
    for (int j = tid; j < SEQ; j += 128) labsh[j] = labels[b * SEQ + j];

    // Stage the 16 anchor rows (16 x 768 f32) into padded LDS
    const float* arow = feat + ((size_t)b * SEQ + iBase) * DIM;
    for (int idx = tid; idx < 16 * (DIM / 4); idx += 128) {
        const int r  = idx / (DIM / 4);
        const int c4 = (idx % (DIM / 4)) * 4;
        const float4 v = *(const float4*)(arow + (size_t)r * DIM + c4);
        float* dst = &As[r * LDA + c4];
        dst[0] = v.x; dst[1] = v.y; dst[2] = v.z; dst[3] = v.w;
    }
    if (tid < 16) {
        negS[tid] = 0.0f; posS[tid] = 0.0f; nLat[tid] = 0.0f;
        const int nn = SEQ - gsize[b * SEQ + iBase + tid];
        rowInv[tid] = 1.0f / (float)(nn < 1 ? 1 : nn);
    }
    __syncthreads();

    // Hoist per-row constants into registers (row r = v + 8*hi for C-fragment VGPR v)
    float invR[8]; int labR[8]; int giR[8];
    #pragma unroll
    for (int v = 0; v < 8; v++) {
        const int r = v + hi * 8;
        giR[v]  = iBase + r;
        labR[v] = labsh[iBase + r];
        invR[v] = rowInv[r];
    }

    const float scale = 1.0f / (0.7f * sqrtf((float)DIM));
    float negA[8], posA[8], cntA[8];
    #pragma unroll
    for (int v = 0; v < 8; v++) { negA[v] = 0.0f; posA[v] = 0.0f; cntA[v] = 0.0f; }

    // A fragment: lane reads As[m=n][k + 2*hi .. +1]; B fragment mirrors with global col row
    const float* aptr  = &As[n * LDA + hi * 2];
    const float* bbase = feat + (size_t)b * SEQ * DIM + (size_t)(hi * 2);

    for (int jt = wave; jt < SEQ / 16; jt += 4) {
        const int jBase = jt * 16;
        const float* bptr = bbase + (size_t)(jBase + n) * DIM;

        v8f c = {};
        #pragma unroll 8
        for (int k = 0; k < DIM; k += 4) {
            const v2f a  = *(const v2f*)(aptr + k);
            const v2f bb = *(const v2f*)(bptr + k);
            // D = A(16x4) * B(4x16) + C, f32 everywhere
            c = __builtin_amdgcn_wmma_f32_16x16x4_f32(false, a, false, bb,
                                                      (short)0, c, false, false);
        }

        const int gj   = jBase + n;
        const int labj = labsh[gj];
        #pragma unroll
        for (int v = 0; v < 8; v++) {
            const float sim   = c[v] * scale;
            const bool  same  = (labR[v] == labj);
            const bool  later = same && (gj > giR[v]);
            negA[v] += same  ? 0.0f : __expf(sim * invR[v]);
            posA[v] += later ? sim  : 0.0f;
            cntA[v] += later ? 1.0f : 0.0f;
        }
    }

    // Reduce across the 16 lanes of each half-wave (xor network stays within halves)
    #pragma unroll
    for (int v = 0; v < 8; v++) {
        float x = negA[v], y = posA[v], z = cntA[v];
        for (int off = 1; off < 16; off <<= 1) {
            x += __shfl_xor(x, off, 32);
            y += __shfl_xor(y, off, 32);
            z += __shfl_xor(z, off, 32);
        }
        if (n == 0) {
            const int r = v + hi * 8;
            atomicAdd(&negS[r], x);
            atomicAdd(&posS[r], y);
            atomicAdd(&nLat[r], z);
        }
    }
    __syncthreads();

    // Finalize 16 anchor rows, reduce to one value, single global atomic per block
    if (tid < 16) {
        const int gi     = iBase + tid;
        const int gsz    = gsize[b * SEQ + gi];
        const int numneg = SEQ - gsz;
        float contrib = 0.0f;
        if (numneg > 0) {
            const float ln = __logf(negS[tid]);
            contrib = -(posS[tid] - nLat[tid] * ln) / (float)gsz;
        }
        contrib *= (1.0f / (float)SEQ);
        for (int off = 1; off < 16; off <<= 1) contrib += __shfl_xor(contrib, off, 32);
        if (tid == 0) atomicAdd(out, contrib);
    }
}

extern "C" void kernel_launch(void* const* d_in, const int* in_sizes, int n_in,
                              void* d_out, int out_size, void* d_ws, size_t ws_size,
                              hipStream_t stream) {
    const float* feat   = (const float*)d_in[0];
    const int*   labels = (const int*)d_in[1];
    float*       out    = (float*)d_out;
    int*         gsize  = (int*)d_ws;   // 32*1024 ints = 128 KB scratch

    zero_out_kernel<<<1, 1, 0, stream>>>(out);
    label_stats_kernel<<<BATCH, 256, 0, stream>>>(labels, gsize);

    dim3 grid(SEQ / 16, BATCH);   // 64 anchor tiles x 32 batches = 2048 blocks
    supcon_main_kernel<<<grid, 128, 0, stream>>>(feat, labels, gsize, out);
}